// gap_3393024163970
// MI455X (gfx1250) — compile-verified
//
#include <hip/hip_runtime.h>
#include <math.h>

// Problem dims (fixed by the reference)
#define BATCH 16
#define NDIM 2048              // rows used by part0 / cols used by part1
#define SDIM 2049              // full matrix side (n+1 == m+1)
#define GAMMA_F 0.5f
#define ROWS_PER_BLOCK 32
#define THREADS 256
#define NWAVES (THREADS / 32)
#define LBUF_STRIDE 2052       // 2049 rounded up (16B multiple)
#define NBUF 4                 // async pipeline: 3 rows in flight + current
#define DEPTH 3

// ---------------------------------------------------------------------------
// Pass 1: gather positive scores and zero the column accumulators.
//   pos0[b,i] = scores[b, i, gt0']   (gt0' = m if gt0 == -1)
//   pos1[b,j] = scores[b, gt1', j]   (gt1' = n if gt1 == -1)
// ---------------------------------------------------------------------------
__global__ __launch_bounds__(THREADS) void gap_prep_kernel(
    const int* __restrict__ gt0, const int* __restrict__ gt1,
    const float* __restrict__ scores,
    float* __restrict__ pos0, float* __restrict__ pos1,
    float* __restrict__ colsum)
{
    int t = blockIdx.x * THREADS + threadIdx.x;
    if (t >= BATCH * NDIM) return;
    int b = t / NDIM;
    int i = t % NDIM;
    const float* sb = scores + (size_t)b * SDIM * SDIM;

    int g0 = gt0[t];
    int c  = (g0 == -1) ? NDIM : g0;          // column of the positive match
    pos0[t] = sb[(size_t)i * SDIM + c];

    int g1 = gt1[t];
    int r  = (g1 == -1) ? NDIM : g1;          // row of the positive match
    pos1[t] = sb[(size_t)r * SDIM + i];

    colsum[t] = 0.0f;                          // re-zeroed every launch
}

// Issue one thread-stripe (8 dwords, columns tid+256k) of a row as async
// global->LDS loads. IOFFSET is added to BOTH the LDS and global addresses
// (CDNA5 ISA async pseudocode), so one LDS-base VGPR + one global-offset
// VGPR covers the whole stripe with immediate offsets.
__device__ __forceinline__ void issue_async_row(unsigned ldsAddr,
                                                unsigned gOffBytes,
                                                const float* sgprBase)
{
    asm volatile(
        "global_load_async_to_lds_b32 %0, %1, %2 offset:0\n\t"
        "global_load_async_to_lds_b32 %0, %1, %2 offset:1024\n\t"
        "global_load_async_to_lds_b32 %0, %1, %2 offset:2048\n\t"
        "global_load_async_to_lds_b32 %0, %1, %2 offset:3072\n\t"
        "global_load_async_to_lds_b32 %0, %1, %2 offset:4096\n\t"
        "global_load_async_to_lds_b32 %0, %1, %2 offset:5120\n\t"
        "global_load_async_to_lds_b32 %0, %1, %2 offset:6144\n\t"
        "global_load_async_to_lds_b32 %0, %1, %2 offset:7168"
        :
        : "v"(ldsAddr), "v"(gOffBytes), "s"(sgprBase)
        : "memory");
}

// Wait until at most `ahead` whole rows (8 async dwords each) remain
// outstanding; in-order completion guarantees all earlier rows landed.
__device__ __forceinline__ void wait_async_rows(int ahead)   // uniform
{
    if (ahead >= DEPTH)      asm volatile("s_wait_asynccnt 0x18" ::: "memory");
    else if (ahead == 2)     asm volatile("s_wait_asynccnt 0x10" ::: "memory");
    else if (ahead == 1)     asm volatile("s_wait_asynccnt 0x8"  ::: "memory");
    else                     asm volatile("s_wait_asynccnt 0x0"  ::: "memory");
}

// ---------------------------------------------------------------------------
// Pass 2: single streaming sweep over scores (269 MB read once), pipelined
// through LDS with async global->LDS loads, 3 rows in flight per wave.
// Block = 256 threads, ROWS_PER_BLOCK rows of one batch. Thread owns columns
// {tid + 256k, k=0..7}; col 2048 (row-sum only) handled by tid 0 directly.
// ---------------------------------------------------------------------------
__global__ __launch_bounds__(THREADS) void gap_main_kernel(
    const float* __restrict__ scores,
    const float* __restrict__ pos0, const float* __restrict__ pos1,
    float* __restrict__ rowsum, float* __restrict__ colsum)
{
    __shared__ float lbuf[NBUF][LBUF_STRIDE];  // 4 x 8.2 KB row buffers
    __shared__ float wsum[2][NWAVES];          // parity-buffered row staging

    const int tid    = threadIdx.x;
    const int b      = blockIdx.y;
    const int row0   = blockIdx.x * ROWS_PER_BLOCK;
    const int rowEnd = (row0 + ROWS_PER_BLOCK < SDIM) ? row0 + ROWS_PER_BLOCK : SDIM;
    const float* sb  = scores + (size_t)b * SDIM * SDIM;

    // LDS byte address of this thread's slot in buffer 0 (shared aperture is
    // in the high 32 bits of the generic address -> low 32 bits = LDS offset).
    const unsigned ldsBase = (unsigned)(uintptr_t)(void*)&lbuf[0][tid];

    float p1v[8], acc[8];
#pragma unroll
    for (int k = 0; k < 8; ++k) {
        p1v[k] = pos1[b * NDIM + tid + 256 * k];
        acc[k] = 0.0f;
    }

    // Prologue: kick off up to DEPTH rows.
#pragma unroll
    for (int d = 0; d < DEPTH; ++d) {
        int r = row0 + d;
        if (r < rowEnd)
            issue_async_row(ldsBase + (unsigned)((r % NBUF) * (LBUF_STRIDE * 4)),
                            (unsigned)(((size_t)r * SDIM + tid) * 4u), sb);
    }

    for (int i = row0; i < rowEnd; ++i) {
        const int cur = i % NBUF;
        const int par = i & 1;
        const bool rowActive = (i < NDIM);
        const float pos = rowActive ? pos0[b * NDIM + i] : 0.0f;

        // Keep DEPTH rows in flight, then wait for the current row.
        int nxt = i + DEPTH;
        if (nxt < rowEnd)
            issue_async_row(ldsBase + (unsigned)((nxt % NBUF) * (LBUF_STRIDE * 4)),
                            (unsigned)(((size_t)nxt * SDIM + tid) * 4u), sb);
        int ahead = rowEnd - 1 - i;            // rows still outstanding after i
        wait_async_rows(ahead < DEPTH ? ahead : DEPTH);

        // Consume the staged row from LDS (conflict-free: per-wave addresses
        // are consecutive for each k).
        float v[8];
#pragma unroll
        for (int k = 0; k < 8; ++k)
            v[k] = lbuf[cur][tid + 256 * k];

        // Column-direction gaps (all 2049 rows contribute; cols < 2048).
#pragma unroll
        for (int k = 0; k < 8; ++k)
            acc[k] += fmaxf(v[k] - p1v[k] + GAMMA_F, 0.0f);

        // Row-direction gap sum (rows < 2048; all 2049 columns contribute).
        if (rowActive) {
            float rp = 0.0f;
#pragma unroll
            for (int k = 0; k < 8; ++k)
                rp += fmaxf(v[k] - pos + GAMMA_F, 0.0f);
            if (tid == 0) {
                float ve = __builtin_nontemporal_load(sb + (size_t)i * SDIM + NDIM);
                rp += fmaxf(ve - pos + GAMMA_F, 0.0f);
            }
            // wave32 reduction
#pragma unroll
            for (int off = 16; off > 0; off >>= 1)
                rp += __shfl_down(rp, off, 32);
            if ((tid & 31) == 0) wsum[par][tid >> 5] = rp;
            __syncthreads();
            if (tid == 0) {
                float tsum = 0.0f;
#pragma unroll
                for (int w = 0; w < NWAVES; ++w) tsum += wsum[par][w];
                rowsum[b * NDIM + i] = tsum;
            }
            // No trailing barrier: next row writes the other parity slot, and
            // its barrier orders the row-after-next's writes past our read.
        }
    }

    // Flush this tile's column partials (65 tiles/batch -> light contention).
#pragma unroll
    for (int k = 0; k < 8; ++k)
        atomicAdd(&colsum[b * NDIM + tid + 256 * k], acc[k]);
}

// ---------------------------------------------------------------------------
// Pass 3: per-batch finalize.
//   loss = 2*log(sum - gamma + 1) = 2*log(sum + 0.5); mean rows + mean cols,
//   quartered -> divide the grand total by 8192.
// ---------------------------------------------------------------------------
__global__ __launch_bounds__(THREADS) void gap_final_kernel(
    const float* __restrict__ rowsum, const float* __restrict__ colsum,
    float* __restrict__ out)
{
    __shared__ float red[NWAVES];
    const int b = blockIdx.x;
    const int tid = threadIdx.x;

    float s = 0.0f;
    for (int k = tid; k < NDIM; k += THREADS)
        s += 2.0f * logf(rowsum[b * NDIM + k] + (1.0f - GAMMA_F));
    for (int k = tid; k < NDIM; k += THREADS)
        s += 2.0f * logf(colsum[b * NDIM + k] + (1.0f - GAMMA_F));

#pragma unroll
    for (int off = 16; off > 0; off >>= 1)
        s += __shfl_down(s, off, 32);
    if ((tid & 31) == 0) red[tid >> 5] = s;
    __syncthreads();
    if (tid == 0) {
        float t = 0.0f;
#pragma unroll
        for (int w = 0; w < NWAVES; ++w) t += red[w];
        out[b] = t / (2.0f * 2.0f * (float)NDIM);   // /8192
    }
}

// ---------------------------------------------------------------------------
extern "C" void kernel_launch(void* const* d_in, const int* in_sizes, int n_in,
                              void* d_out, int out_size, void* d_ws, size_t ws_size,
                              hipStream_t stream)
{
    (void)in_sizes; (void)n_in; (void)out_size; (void)ws_size;

    const int*   gt0    = (const int*)d_in[0];     // gt_matches0 (16,2048)
    const int*   gt1    = (const int*)d_in[1];     // gt_matches1 (16,2048)
    const float* scores = (const float*)d_in[2];   // (16,2049,2049)
    float*       out    = (float*)d_out;           // (16,)

    // Workspace layout (floats): 4 * 16*2048 = 512 KB total.
    float* ws     = (float*)d_ws;
    float* pos0   = ws;                    // 32768
    float* pos1   = ws + 32768;            // 32768
    float* rowsum = ws + 65536;            // 32768
    float* colsum = ws + 98304;            // 32768

    const int totPrep = BATCH * NDIM;
    gap_prep_kernel<<<(totPrep + THREADS - 1) / THREADS, THREADS, 0, stream>>>(
        gt0, gt1, scores, pos0, pos1, colsum);

    dim3 grid((SDIM + ROWS_PER_BLOCK - 1) / ROWS_PER_BLOCK, BATCH);
    gap_main_kernel<<<grid, THREADS, 0, stream>>>(scores, pos0, pos1, rowsum, colsum);

    gap_final_kernel<<<BATCH, THREADS, 0, stream>>>(rowsum, colsum, out);
}